// ChamferDistance_11012296147710
// MI455X (gfx1250) — compile-verified
//
#include <hip/hip_runtime.h>
#include <math.h>

typedef __attribute__((ext_vector_type(2))) float v2f;
typedef __attribute__((ext_vector_type(8))) float v8f;

#define BATCH 8
#define NPTS 4096
#define MTILES 4           // i-tiles (of 16 rows) per wave -> 64 rows/wave
#define WAVES_PER_BLOCK 4  // 128-thread blocks = 4 wave32

// ---------------------------------------------------------------------------
// Kernel 1: initialize dist2 (column mins) to +inf bit pattern.
// Run every launch so graph replay is deterministic.
// ---------------------------------------------------------------------------
__global__ void cd_init_kernel(unsigned int* __restrict__ dist2bits) {
  int i = blockIdx.x * blockDim.x + threadIdx.x;
  if (i < BATCH * NPTS) dist2bits[i] = 0x7F800000u;  // +inf
}

// ---------------------------------------------------------------------------
// Kernel 2: fused pairwise-distance + dual min reduction via WMMA f32 16x16x4.
// Each wave: batch b, a 64-row block of pcs1; loops over all 256 j-tiles.
// ---------------------------------------------------------------------------
__global__ __launch_bounds__(32 * WAVES_PER_BLOCK)
void cd_main_kernel(const float* __restrict__ p1, const float* __restrict__ p2,
                    float* __restrict__ dist1,
                    unsigned int* __restrict__ dist2bits) {
  const int lane = threadIdx.x & 31;
  const int wave = threadIdx.x >> 5;
  const int gw   = blockIdx.x * WAVES_PER_BLOCK + wave;   // 0..511
  const int b    = gw >> 6;                                // 64 i-blocks/batch
  const int iblk = gw & 63;
  const int i0   = iblk * (16 * MTILES);
  const int hi   = lane >> 4;   // 0: lanes 0-15 (K=0,1), 1: lanes 16-31 (K=2,3)
  const int m    = lane & 15;

  const float* __restrict__ p1b = p1 + (size_t)b * NPTS * 3;
  const float* __restrict__ p2b = p2 + (size_t)b * NPTS * 3;

  // ---- Setup: A matrices (16x4 f32 layout) and row norms, held in VGPRs ----
  // A layout (ISA 7.12.2): lanes 0-15: M=lane, V0=K0,V1=K1; lanes 16-31: K2,K3.
  v2f   Am[MTILES];
  float s1[MTILES][8];
  for (int t = 0; t < MTILES; ++t) {
    const int row = i0 + t * 16 + m;
    const float x = p1b[3 * row + 0];
    const float y = p1b[3 * row + 1];
    const float z = p1b[3 * row + 2];
    Am[t].x = hi ? z : x;       // K=0 (low) / K=2 (high)
    Am[t].y = hi ? 0.0f : y;    // K=1 (low) / K=3 pad (high)
    // Row norms for the D-tile rows this lane's VGPRs correspond to:
    // D[M,N]: VGPR v -> row M=v (lanes 0-15) or M=v+8 (lanes 16-31).
    for (int v = 0; v < 8; ++v) {
      const int r  = i0 + t * 16 + v + 8 * hi;
      const float xx = p1b[3 * r + 0];
      const float yy = p1b[3 * r + 1];
      const float zz = p1b[3 * r + 2];
      s1[t][v] = xx * xx + yy * yy + zz * zz;
    }
  }

  float rowMin[MTILES][8];
  for (int t = 0; t < MTILES; ++t)
    for (int v = 0; v < 8; ++v) rowMin[t][v] = 3.4e38f;

  // ---- Main loop over the 256 column tiles of pcs2 ----
  for (int jt = 0; jt < NPTS / 16; ++jt) {
    const int col = jt * 16 + m;
    const float x = p2b[3 * col + 0];
    const float y = p2b[3 * col + 1];
    const float z = p2b[3 * col + 2];
    const float sq2 = x * x + y * y + z * z;

    // B matrix 4x16 f32 layout mirrors A: low lanes (x,y)=K0,K1; high (z,0)=K2,K3.
    v2f Bm;
    Bm.x = hi ? z : x;
    Bm.y = hi ? 0.0f : y;

    float cmin = 3.4e38f;
    for (int t = 0; t < MTILES; ++t) {
      v8f acc = {};
      acc = __builtin_amdgcn_wmma_f32_16x16x4_f32(
          /*neg_a=*/false, Am[t], /*neg_b=*/false, Bm,
          /*c_mod=*/(short)0, acc, /*reuse_a=*/false, /*reuse_b=*/false);
      // acc[v] = dot(p1[i0+t*16+v+8*hi], p2[jt*16+m])
      for (int v = 0; v < 8; ++v) {
        float d = fmaf(-2.0f, acc[v], s1[t][v] + sq2);
        d = fmaxf(d, 0.0f);
        rowMin[t][v] = fminf(rowMin[t][v], d);
        cmin = fminf(cmin, d);
      }
    }
    // Column min across this wave's 64 rows -> global (uint-bit min is valid
    // float min for non-negative values). Lanes L and L+16 share a column.
    atomicMin(&dist2bits[b * NPTS + col], __float_as_uint(cmin));
  }

  // ---- dist1: reduce rowMin across the 16 lanes of each half-wave ----
  // XOR masks 1,2,4,8 never cross bit 4, so the two 16-lane halves (which hold
  // different row sets) reduce independently.
  for (int t = 0; t < MTILES; ++t) {
    for (int v = 0; v < 8; ++v) {
      float r = rowMin[t][v];
      for (int mask = 1; mask < 16; mask <<= 1)
        r = fminf(r, __shfl_xor(r, mask, 32));
      if (m == 0) dist1[b * NPTS + i0 + t * 16 + v + 8 * hi] = r;
    }
  }
}

// ---------------------------------------------------------------------------
// Kernel 3: loss = 0.5*(mean(sqrt(dist1)) + mean(sqrt(dist2)))
//                = (sum sqrt(dist1) + sum sqrt(dist2)) / (2*B*N)
// ---------------------------------------------------------------------------
__global__ void cd_reduce_kernel(const float* __restrict__ dist1,
                                 const unsigned int* __restrict__ dist2bits,
                                 float* __restrict__ out) {
  __shared__ float sm[256];
  float acc = 0.0f;
  for (int i = threadIdx.x; i < BATCH * NPTS; i += 256)
    acc += sqrtf(dist1[i]);
  for (int i = threadIdx.x; i < BATCH * NPTS; i += 256)
    acc += sqrtf(__uint_as_float(dist2bits[i]));
  sm[threadIdx.x] = acc;
  __syncthreads();
  for (int s = 128; s > 0; s >>= 1) {
    if (threadIdx.x < (unsigned)s) sm[threadIdx.x] += sm[threadIdx.x + s];
    __syncthreads();
  }
  if (threadIdx.x == 0)
    out[0] = sm[0] / (float)(2 * BATCH * NPTS);
}

// ---------------------------------------------------------------------------
extern "C" void kernel_launch(void* const* d_in, const int* in_sizes, int n_in,
                              void* d_out, int out_size, void* d_ws, size_t ws_size,
                              hipStream_t stream) {
  const float* p1 = (const float*)d_in[0];  // [8,4096,3] f32
  const float* p2 = (const float*)d_in[1];  // [8,4096,3] f32

  // Workspace layout: dist1 floats [0, B*N), dist2 uint bits [B*N, 2*B*N)
  float*        dist1     = (float*)d_ws;
  unsigned int* dist2bits = (unsigned int*)d_ws + BATCH * NPTS;

  cd_init_kernel<<<(BATCH * NPTS + 255) / 256, 256, 0, stream>>>(dist2bits);

  const int totalWaves = BATCH * (NPTS / (16 * MTILES));  // 512
  cd_main_kernel<<<totalWaves / WAVES_PER_BLOCK, 32 * WAVES_PER_BLOCK, 0, stream>>>(
      p1, p2, dist1, dist2bits);

  cd_reduce_kernel<<<1, 256, 0, stream>>>(dist1, dist2bits, (float*)d_out);
}